// GraphAttentionNetwork_61168924230411
// MI455X (gfx1250) — compile-verified
//
#include <hip/hip_runtime.h>
#include <math.h>

#define BB 64
#define TT 512
#define HIDD 8192
#define CC 64
#define NHH 4
#define HDD 16
#define GDD 64
#define INDIM 128

typedef float v2f __attribute__((ext_vector_type(2)));
typedef float v4f __attribute__((ext_vector_type(4)));
typedef float v8f __attribute__((ext_vector_type(8)));

// ---------------------------------------------------------------------------
// 1) Temporal mean pool: node_inputs[b, x] = mean_t H[b, t, x]   (x = c*128+f)
//    1 GiB streamed once -> non-temporal b128 loads, fully coalesced.
//    grid = (HID/1024, B), block = 256, each thread owns 4 floats.
// ---------------------------------------------------------------------------
__global__ void pool_mean_kernel(const float* __restrict__ H,
                                 float* __restrict__ ni) {
    const int b    = blockIdx.y;
    const int base = blockIdx.x * 1024 + threadIdx.x * 4;
    const float* p = H + (size_t)b * TT * HIDD + base;
    v4f acc = {0.f, 0.f, 0.f, 0.f};
#pragma unroll 8
    for (int t = 0; t < TT; ++t) {
        v4f v = __builtin_nontemporal_load((const v4f*)(p + (size_t)t * HIDD));
        acc += v;
    }
    const float s = 1.0f / (float)TT;
    v4f r = acc * s;
    *(v4f*)(ni + (size_t)b * HIDD + base) = r;
}

// ---------------------------------------------------------------------------
// 2) Flat GEMM via V_WMMA_F32_16X16X4_F32:
//    Out[m, n] = sum_k A[m, k] * Bm[n, k]  (+ bias[n] if given)
//    M = 4096, N = 64.  One wave per 16x16 output tile -> 1024 waves.
//    f32 WMMA fragment layout (ISA 7.12.2):
//      A 16x4 : lane<16 -> row=m0+lane, (K,K+1); lane>=16 -> (K+2,K+3)
//      B 4x16 : lane<16 -> col=n0+lane, (K,K+1); lane>=16 -> (K+2,K+3)
//      C/D    : acc[v] -> row = m0 + v + 8*(lane>=16), col = n0 + (lane&15)
//    Two accumulators break the WMMA->WMMA RAW chain.
// ---------------------------------------------------------------------------
__global__ void gemm_nt_wmma_kernel(const float* __restrict__ A,
                                    const float* __restrict__ Bm,
                                    const float* __restrict__ bias,
                                    float* __restrict__ Out,
                                    int K) {
    const int wave = (blockIdx.x * blockDim.x + threadIdx.x) >> 5;
    const int lane = threadIdx.x & 31;
    const int nt = wave & 3;                 // N tiles: 64/16 = 4
    const int mt = wave >> 2;                // M tiles: 4096/16 = 256
    const int m0 = mt * 16, n0 = nt * 16;
    const int half = lane >> 4;              // 0: K,K+1   1: K+2,K+3
    const int lm   = lane & 15;

    const float* arow = A  + (size_t)(m0 + lm) * K + 2 * half;
    const float* brow = Bm + (size_t)(n0 + lm) * K + 2 * half;

    v8f acc0 = {0.f,0.f,0.f,0.f,0.f,0.f,0.f,0.f};
    v8f acc1 = {0.f,0.f,0.f,0.f,0.f,0.f,0.f,0.f};
    for (int k = 0; k < K; k += 8) {
        v2f a0, b0, a1, b1;
        a0.x = arow[k];     a0.y = arow[k + 1];
        b0.x = brow[k];     b0.y = brow[k + 1];
        a1.x = arow[k + 4]; a1.y = arow[k + 5];
        b1.x = brow[k + 4]; b1.y = brow[k + 5];
        acc0 = __builtin_amdgcn_wmma_f32_16x16x4_f32(false, a0, false, b0,
                                                     (short)0, acc0, false, false);
        acc1 = __builtin_amdgcn_wmma_f32_16x16x4_f32(false, a1, false, b1,
                                                     (short)0, acc1, false, false);
    }
    v8f acc = acc0 + acc1;

    const float bval = bias ? bias[n0 + lm] : 0.0f;
    float* out = Out + n0 + lm;
#pragma unroll
    for (int v = 0; v < 8; ++v) {
        const int row = m0 + v + 8 * half;
        out[(size_t)row * GDD] = acc[v] + bval;
    }
}

// ---------------------------------------------------------------------------
// 3) Attention per (b, h): scores, leaky_relu(0.2), softmax over j,
//    attn -> d_out; head_out = elu(attn @ Wh); column mean over c -> nmean.
//    grid = B*NH, block = 64 (thread i owns row i).
// ---------------------------------------------------------------------------
__global__ void attn_kernel(const float* __restrict__ WhAll,  // [B*C, 64], n = h*16+o
                            const float* __restrict__ aw,     // [NH, 32]
                            float* __restrict__ attn_out,     // [B, NH, C, C]
                            float* __restrict__ nmean) {      // [B, 64]
    const int b = blockIdx.x >> 2;
    const int h = blockIdx.x & 3;
    const int i = threadIdx.x;               // 0..63 (node / row index)

    __shared__ float Wh[CC][HDD];
    __shared__ float s1[CC], s2[CC];
    __shared__ float ex[CC][CC];
    __shared__ float ho[CC][HDD];

#pragma unroll
    for (int o = 0; o < HDD; ++o)
        Wh[i][o] = WhAll[(size_t)(b * CC + i) * GDD + h * HDD + o];
    __syncthreads();

    {
        float d1 = 0.f, d2 = 0.f;
#pragma unroll
        for (int o = 0; o < HDD; ++o) {
            const float w = Wh[i][o];
            d1 += w * aw[h * 2 * HDD + o];
            d2 += w * aw[h * 2 * HDD + HDD + o];
        }
        s1[i] = d1; s2[i] = d2;
    }
    __syncthreads();

    const float si = s1[i];
    float m = -INFINITY;
    for (int j = 0; j < CC; ++j) {
        float e = si + s2[j];
        e = (e > 0.f) ? e : 0.2f * e;        // leaky_relu(0.2)
        m = fmaxf(m, e);
    }
    float den = 0.f;
    for (int j = 0; j < CC; ++j) {
        float e = si + s2[j];
        e = (e > 0.f) ? e : 0.2f * e;
        const float ee = expf(e - m);
        ex[i][j] = ee;
        den += ee;
    }
    const float inv = 1.0f / den;

    float* arow = attn_out + ((size_t)(b * NHH + h) * CC + i) * CC;
    for (int j = 0; j < CC; ++j) arow[j] = ex[i][j] * inv;

    float acc[HDD];
#pragma unroll
    for (int o = 0; o < HDD; ++o) acc[o] = 0.f;
    for (int j = 0; j < CC; ++j) {
        const float a = ex[i][j] * inv;
#pragma unroll
        for (int o = 0; o < HDD; ++o) acc[o] += a * Wh[j][o];
    }
#pragma unroll
    for (int o = 0; o < HDD; ++o) {
        const float v = acc[o];
        ho[i][o] = (v > 0.f) ? v : expm1f(v);   // elu
    }
    __syncthreads();

    if (i < HDD) {
        float s = 0.f;
        for (int c = 0; c < CC; ++c) s += ho[c][i];
        nmean[b * GDD + h * HDD + i] = s * (1.0f / (float)CC);
    }
}

// ---------------------------------------------------------------------------
// 4) Readout: g[b, i] = relu(nmean[b, :] . Wr[i, :] + br[i])
// ---------------------------------------------------------------------------
__global__ void readout_kernel(const float* __restrict__ nmean,
                               const float* __restrict__ Wr,
                               const float* __restrict__ br,
                               float* __restrict__ g) {
    const int b = blockIdx.x;
    const int i = threadIdx.x;               // 0..127
    float acc = br[i];
    for (int k = 0; k < GDD; ++k)
        acc += nmean[b * GDD + k] * Wr[i * GDD + k];
    g[b * INDIM + i] = fmaxf(acc, 0.0f);
}

// ---------------------------------------------------------------------------
extern "C" void kernel_launch(void* const* d_in, const int* in_sizes, int n_in,
                              void* d_out, int out_size, void* d_ws, size_t ws_size,
                              hipStream_t stream) {
    const float* H    = (const float*)d_in[0];   // [64, 512, 8192]
    const float* We   = (const float*)d_in[1];   // [64, 128]
    const float* be   = (const float*)d_in[2];   // [64]
    const float* Whw  = (const float*)d_in[3];   // [4, 16, 64] -> viewed as [64(n)][64(g)]
    const float* aw   = (const float*)d_in[4];   // [4, 32]
    const float* Wr   = (const float*)d_in[5];   // [128, 64]
    const float* br   = (const float*)d_in[6];   // [128]

    float* out_g    = (float*)d_out;             // [64, 128]  (8192 floats)
    float* out_attn = (float*)d_out + BB * INDIM;// [64, 4, 64, 64]

    float* ws    = (float*)d_ws;
    float* ni    = ws;                           // [4096, 128]  node_inputs
    float* nodes = ni    + (size_t)BB * HIDD;    // [4096, 64]
    float* WhAll = nodes + (size_t)BB * CC * GDD;// [4096, 64]   heads concat in N
    float* nmean = WhAll + (size_t)BB * CC * GDD;// [64, 64]

    // 1) mean pool over T  (bandwidth-dominant: ~1 GiB NT stream)
    pool_mean_kernel<<<dim3(HIDD / 1024, BB), 256, 0, stream>>>(H, ni);

    // 2) node embedding: [4096,128] x [128,64]^T + bias  (WMMA f32)
    gemm_nt_wmma_kernel<<<256, 128, 0, stream>>>(ni, We, be, nodes, INDIM);

    // 3) per-head projection: [4096,64] x [64,64]^T  (WMMA f32)
    //    Bm row n = h*16+o is exactly Whw[h, o, :] in flat layout.
    gemm_nt_wmma_kernel<<<256, 128, 0, stream>>>(nodes, Whw, nullptr, WhAll, GDD);

    // 4) attention + softmax + elu + column mean
    attn_kernel<<<BB * NHH, CC, 0, stream>>>(WhAll, aw, out_attn, nmean);

    // 5) readout
    readout_kernel<<<BB, INDIM, 0, stream>>>(nmean, Wr, br, out_g);
}